// WeaveLayer_1082331758607
// MI455X (gfx1250) — compile-verified
//
#include <hip/hip_runtime.h>

// Fused WeaveLayer for gfx1250 (MI455X), fp32 throughout via V_WMMA_F32_16X16X4_F32.
// ws layout (floats): [0,V*H) edge_node accum, [V*H,2V*H) left, [2V*H,3V*H) right  (~60MB).
// Streaming traffic (edge_feats in, new_edge out) uses non-temporal hints so the
// L2-resident gather/scatter tables are not evicted.

typedef __attribute__((ext_vector_type(2))) float v2f;
typedef __attribute__((ext_vector_type(8))) float v8f;

#define HDIM 50
#define NIN  128
#define KE   52    // padded K for edge_feats GEMMs (50 -> 52, 13 wmma steps)
#define KC   152   // padded K for [first|second|third] GEMM (150 -> 152, 38 steps)
#define KC2  104   // padded K for [node_node|edge_node] GEMM (100 -> 104, 26 steps)

static __device__ __forceinline__ v8f wmma_f32(v2f a, v2f b, v8f c) {
  // D = A(16x4,f32) * B(4x16,f32) + C(16x16,f32); NEG/NEG_HI must be 0 for f32 A/B.
  return __builtin_amdgcn_wmma_f32_16x16x4_f32(false, a, false, b, (short)0, c, false, false);
}
static __device__ __forceinline__ float relu(float x) { return fmaxf(x, 0.0f); }

__global__ void zero_kernel(float* __restrict__ p, long n) {
  long i = (long)blockIdx.x * blockDim.x + threadIdx.x;
  if (i < n) p[i] = 0.0f;
}

// ---------------------------------------------------------------------------
// left = nf @ w_l + b_l ; right = nf @ w_r + b_r   (V x 50), no relu.
// One wave = 16 node rows. K=128 -> 32 wmma steps, 4 column tiles of 16.
// ---------------------------------------------------------------------------
__global__ __launch_bounds__(256) void node_lr_kernel(
    const float* __restrict__ nf,
    const float* __restrict__ w_l, const float* __restrict__ b_l,
    const float* __restrict__ w_r, const float* __restrict__ b_r,
    float* __restrict__ left, float* __restrict__ right, int V) {
  __shared__ float wtL[64][NIN];   // w_l^T, rows >=50 zero
  __shared__ float wtR[64][NIN];
  __shared__ float bL[64], bR[64];
  const int tid = threadIdx.x;
  for (int i = tid; i < 64 * NIN; i += 256) {
    int n = i / NIN, k = i % NIN;
    wtL[n][k] = (n < HDIM) ? w_l[k * HDIM + n] : 0.0f;
    wtR[n][k] = (n < HDIM) ? w_r[k * HDIM + n] : 0.0f;
  }
  if (tid < 64) { bL[tid] = tid < HDIM ? b_l[tid] : 0.0f; bR[tid] = tid < HDIM ? b_r[tid] : 0.0f; }
  __syncthreads();

  const int wave = tid >> 5, lane = tid & 31, half = lane >> 4, mrow = lane & 15;
  const int v0 = (blockIdx.x * 8 + wave) * 16;
  if (v0 >= V) return;

  const float* arow = nf + (size_t)(v0 + mrow) * NIN;
  v2f af[32];
#pragma unroll
  for (int s = 0; s < 32; ++s) af[s] = *(const v2f*)(arow + s * 4 + 2 * half);

  v8f acc[4];
#pragma unroll
  for (int t = 0; t < 4; ++t) acc[t] = (v8f){0.f,0.f,0.f,0.f,0.f,0.f,0.f,0.f};
#pragma unroll
  for (int s = 0; s < 32; ++s) {
#pragma unroll
    for (int t = 0; t < 4; ++t)
      acc[t] = wmma_f32(af[s], *(const v2f*)&wtL[t * 16 + mrow][s * 4 + 2 * half], acc[t]);
  }
#pragma unroll
  for (int t = 0; t < 4; ++t) {
    int col = t * 16 + mrow;
    if (col < HDIM) {
#pragma unroll
      for (int g = 0; g < 8; ++g)
        left[(size_t)(v0 + g + 8 * half) * HDIM + col] = acc[t][g] + bL[col];
    }
  }
#pragma unroll
  for (int t = 0; t < 4; ++t) acc[t] = (v8f){0.f,0.f,0.f,0.f,0.f,0.f,0.f,0.f};
#pragma unroll
  for (int s = 0; s < 32; ++s) {
#pragma unroll
    for (int t = 0; t < 4; ++t)
      acc[t] = wmma_f32(af[s], *(const v2f*)&wtR[t * 16 + mrow][s * 4 + 2 * half], acc[t]);
  }
#pragma unroll
  for (int t = 0; t < 4; ++t) {
    int col = t * 16 + mrow;
    if (col < HDIM) {
#pragma unroll
      for (int g = 0; g < 8; ++g)
        right[(size_t)(v0 + g + 8 * half) * HDIM + col] = acc[t][g] + bR[col];
    }
  }
}

// ---------------------------------------------------------------------------
// Fused edge pipeline. One wave = 16 edges.
//   msg   = relu(ef @ w_e2n + b)  -> atomic scatter-add into edge_node[dst]
//   third = relu(ef @ w_e2e + b)  -> LDS staging cols 100..149
//   first/second via gathers from left/right tables -> staging cols 0..99
//   new_edge = relu([first|second|third] @ w_upd_e + b)   (non-temporal store)
// ---------------------------------------------------------------------------
__global__ __launch_bounds__(256) void edge_kernel(
    const float* __restrict__ ef, const int* __restrict__ src, const int* __restrict__ dst,
    const float* __restrict__ left, const float* __restrict__ right,
    const float* __restrict__ w_e2n, const float* __restrict__ b_e2n,
    const float* __restrict__ w_e2e, const float* __restrict__ b_e2e,
    const float* __restrict__ w_upd, const float* __restrict__ b_upd,
    float* __restrict__ eacc, float* __restrict__ new_edge, int E) {
  __shared__ float wtM[64][KE];           // w_e2n^T zero-padded
  __shared__ float wtT[64][KE];           // w_e2e^T
  __shared__ float wtU[64][KC];           // w_upd_e^T
  __shared__ float bM[64], bT[64], bU[64];
  __shared__ float stage[8][16][KC];      // per-wave 16x152 A-tile for final GEMM
  const int tid = threadIdx.x;
  for (int i = tid; i < 64 * KE; i += 256) {
    int n = i / KE, k = i % KE;
    wtM[n][k] = (n < HDIM && k < HDIM) ? w_e2n[k * HDIM + n] : 0.0f;
    wtT[n][k] = (n < HDIM && k < HDIM) ? w_e2e[k * HDIM + n] : 0.0f;
  }
  for (int i = tid; i < 64 * KC; i += 256) {
    int n = i / KC, k = i % KC;
    wtU[n][k] = (n < HDIM && k < 3 * HDIM) ? w_upd[k * HDIM + n] : 0.0f;
  }
  if (tid < 64) {
    bM[tid] = tid < HDIM ? b_e2n[tid] : 0.0f;
    bT[tid] = tid < HDIM ? b_e2e[tid] : 0.0f;
    bU[tid] = tid < HDIM ? b_upd[tid] : 0.0f;
  }
  __syncthreads();

  const int wave = tid >> 5, lane = tid & 31, half = lane >> 4, mrow = lane & 15;
  const int e0 = (blockIdx.x * 8 + wave) * 16;
  if (e0 >= E) return;

  // A fragments of the 16x50 edge tile (K padded to 52 with zeros).
  // edge_feats is streamed once -> non-temporal loads keep L2 for the tables.
  const float* arow = ef + (size_t)(e0 + mrow) * HDIM;
  v2f af[13];
#pragma unroll
  for (int s = 0; s < 13; ++s) {
    int k0 = s * 4 + 2 * half;
    if (k0 < HDIM) af[s] = __builtin_nontemporal_load((const v2f*)(arow + k0));
    else { af[s].x = 0.0f; af[s].y = 0.0f; }
  }

  v8f acc[4];
  // ---- msg = relu(ef @ w_e2n + b) -> atomic segment_sum ----
#pragma unroll
  for (int t = 0; t < 4; ++t) acc[t] = (v8f){0.f,0.f,0.f,0.f,0.f,0.f,0.f,0.f};
#pragma unroll
  for (int s = 0; s < 13; ++s) {
#pragma unroll
    for (int t = 0; t < 4; ++t)
      acc[t] = wmma_f32(af[s], *(const v2f*)&wtM[t * 16 + mrow][s * 4 + 2 * half], acc[t]);
  }
  int dstm[8];
#pragma unroll
  for (int g = 0; g < 8; ++g) dstm[g] = dst[e0 + g + 8 * half];
#pragma unroll
  for (int t = 0; t < 4; ++t) {
    int col = t * 16 + mrow;
    if (col < HDIM) {
#pragma unroll
      for (int g = 0; g < 8; ++g)
        atomicAdd(&eacc[(size_t)dstm[g] * HDIM + col], relu(acc[t][g] + bM[col]));
    }
  }

  // ---- third = relu(ef @ w_e2e + b) -> staging cols 100..149 ----
#pragma unroll
  for (int t = 0; t < 4; ++t) acc[t] = (v8f){0.f,0.f,0.f,0.f,0.f,0.f,0.f,0.f};
#pragma unroll
  for (int s = 0; s < 13; ++s) {
#pragma unroll
    for (int t = 0; t < 4; ++t)
      acc[t] = wmma_f32(af[s], *(const v2f*)&wtT[t * 16 + mrow][s * 4 + 2 * half], acc[t]);
  }
#pragma unroll
  for (int t = 0; t < 4; ++t) {
    int col = t * 16 + mrow;
    if (col < HDIM) {
#pragma unroll
      for (int g = 0; g < 8; ++g)
        stage[wave][g + 8 * half][2 * HDIM + col] = relu(acc[t][g] + bT[col]);
    }
  }

  // ---- first/second via vectorized gathers (tables are L2-resident) ----
  {
    int e = e0 + mrow;
    int si = src[e], di = dst[e];
    if (half == 0) {
      const float* Ls = left  + (size_t)si * HDIM;   // 200B rows -> 8B aligned
      const float* Rd = right + (size_t)di * HDIM;
#pragma unroll
      for (int c = 0; c < HDIM; c += 2) {
        v2f a = *(const v2f*)(Ls + c);
        v2f b = *(const v2f*)(Rd + c);
        v2f r; r.x = relu(a.x + b.x); r.y = relu(a.y + b.y);
        *(v2f*)&stage[wave][mrow][c] = r;
      }
    } else {
      const float* Rs = right + (size_t)si * HDIM;
      const float* Ld = left  + (size_t)di * HDIM;
#pragma unroll
      for (int c = 0; c < HDIM; c += 2) {
        v2f a = *(const v2f*)(Rs + c);
        v2f b = *(const v2f*)(Ld + c);
        v2f r; r.x = relu(a.x + b.x); r.y = relu(a.y + b.y);
        *(v2f*)&stage[wave][mrow][HDIM + c] = r;
      }
      v2f z; z.x = 0.0f; z.y = 0.0f;
      *(v2f*)&stage[wave][mrow][150] = z;
    }
  }
  asm volatile("" ::: "memory");  // keep LDS fills before LDS reads; DS ops are in-order per wave

  // ---- new_edge = relu(stage(16x152) @ w_upd_e + b) ----
  v8f au[4];
#pragma unroll
  for (int t = 0; t < 4; ++t) au[t] = (v8f){0.f,0.f,0.f,0.f,0.f,0.f,0.f,0.f};
  for (int s = 0; s < 38; ++s) {
    v2f a = *(const v2f*)&stage[wave][mrow][s * 4 + 2 * half];
#pragma unroll
    for (int t = 0; t < 4; ++t)
      au[t] = wmma_f32(a, *(const v2f*)&wtU[t * 16 + mrow][s * 4 + 2 * half], au[t]);
  }
#pragma unroll
  for (int t = 0; t < 4; ++t) {
    int col = t * 16 + mrow;
    if (col < HDIM) {
#pragma unroll
      for (int g = 0; g < 8; ++g)
        __builtin_nontemporal_store(relu(au[t][g] + bU[col]),
                                    &new_edge[(size_t)(e0 + g + 8 * half) * HDIM + col]);
    }
  }
}

// ---------------------------------------------------------------------------
// new_node = relu([relu(nf@w_n2n+b) | edge_node] @ w_upd_n + b)
// ---------------------------------------------------------------------------
__global__ __launch_bounds__(256) void node_update_kernel(
    const float* __restrict__ nf,
    const float* __restrict__ w_n2n, const float* __restrict__ b_n2n,
    const float* __restrict__ w_upd, const float* __restrict__ b_upd,
    const float* __restrict__ eacc, float* __restrict__ new_node, int V) {
  __shared__ float wtN[64][NIN];
  __shared__ float wtU[64][KC2];
  __shared__ float bN[64], bU[64];
  __shared__ float stage[8][16][KC2];
  const int tid = threadIdx.x;
  for (int i = tid; i < 64 * NIN; i += 256) {
    int n = i / NIN, k = i % NIN;
    wtN[n][k] = (n < HDIM) ? w_n2n[k * HDIM + n] : 0.0f;
  }
  for (int i = tid; i < 64 * KC2; i += 256) {
    int n = i / KC2, k = i % KC2;
    wtU[n][k] = (n < HDIM && k < 2 * HDIM) ? w_upd[k * HDIM + n] : 0.0f;
  }
  if (tid < 64) { bN[tid] = tid < HDIM ? b_n2n[tid] : 0.0f; bU[tid] = tid < HDIM ? b_upd[tid] : 0.0f; }
  __syncthreads();

  const int wave = tid >> 5, lane = tid & 31, half = lane >> 4, mrow = lane & 15;
  const int v0 = (blockIdx.x * 8 + wave) * 16;
  if (v0 >= V) return;

  const float* arow = nf + (size_t)(v0 + mrow) * NIN;
  v2f af[32];
#pragma unroll
  for (int s = 0; s < 32; ++s) af[s] = *(const v2f*)(arow + s * 4 + 2 * half);

  v8f acc[4];
#pragma unroll
  for (int t = 0; t < 4; ++t) acc[t] = (v8f){0.f,0.f,0.f,0.f,0.f,0.f,0.f,0.f};
#pragma unroll
  for (int s = 0; s < 32; ++s) {
#pragma unroll
    for (int t = 0; t < 4; ++t)
      acc[t] = wmma_f32(af[s], *(const v2f*)&wtN[t * 16 + mrow][s * 4 + 2 * half], acc[t]);
  }
#pragma unroll
  for (int t = 0; t < 4; ++t) {
    int col = t * 16 + mrow;
    if (col < HDIM) {
#pragma unroll
      for (int g = 0; g < 8; ++g)
        stage[wave][g + 8 * half][col] = relu(acc[t][g] + bN[col]);
    }
  }
  {
    // copy edge_node row (25 aligned float2s split across the two half-waves)
    const float* en = eacc + (size_t)(v0 + mrow) * HDIM;
    int j0 = (half == 0) ? 0 : 13;
    int j1 = (half == 0) ? 13 : 25;
    for (int j = j0; j < j1; ++j)
      *(v2f*)&stage[wave][mrow][HDIM + 2 * j] = *(const v2f*)(en + 2 * j);
    v2f z; z.x = 0.0f; z.y = 0.0f;
    *(v2f*)&stage[wave][mrow][100 + 2 * half] = z;
  }
  asm volatile("" ::: "memory");

  v8f au[4];
#pragma unroll
  for (int t = 0; t < 4; ++t) au[t] = (v8f){0.f,0.f,0.f,0.f,0.f,0.f,0.f,0.f};
  for (int s = 0; s < 26; ++s) {
    v2f a = *(const v2f*)&stage[wave][mrow][s * 4 + 2 * half];
#pragma unroll
    for (int t = 0; t < 4; ++t)
      au[t] = wmma_f32(a, *(const v2f*)&wtU[t * 16 + mrow][s * 4 + 2 * half], au[t]);
  }
#pragma unroll
  for (int t = 0; t < 4; ++t) {
    int col = t * 16 + mrow;
    if (col < HDIM) {
#pragma unroll
      for (int g = 0; g < 8; ++g)
        new_node[(size_t)(v0 + g + 8 * half) * HDIM + col] = relu(au[t][g] + bU[col]);
    }
  }
}

extern "C" void kernel_launch(void* const* d_in, const int* in_sizes, int n_in,
                              void* d_out, int out_size, void* d_ws, size_t ws_size,
                              hipStream_t stream) {
  (void)n_in; (void)out_size; (void)ws_size;
  const float* nf      = (const float*)d_in[0];
  const float* ef      = (const float*)d_in[1];
  const int*   src     = (const int*)d_in[2];
  const int*   dst     = (const int*)d_in[3];
  const float* w_n2n   = (const float*)d_in[4];
  const float* b_n2n   = (const float*)d_in[5];
  const float* w_e2n   = (const float*)d_in[6];
  const float* b_e2n   = (const float*)d_in[7];
  const float* w_upd_n = (const float*)d_in[8];
  const float* b_upd_n = (const float*)d_in[9];
  const float* w_l     = (const float*)d_in[10];
  const float* b_l     = (const float*)d_in[11];
  const float* w_r     = (const float*)d_in[12];
  const float* b_r     = (const float*)d_in[13];
  const float* w_e2e   = (const float*)d_in[14];
  const float* b_e2e   = (const float*)d_in[15];
  const float* w_upd_e = (const float*)d_in[16];
  const float* b_upd_e = (const float*)d_in[17];

  const int V = in_sizes[0] / NIN;
  const int E = in_sizes[2];

  float* out      = (float*)d_out;
  float* new_node = out;
  float* new_edge = out + (size_t)V * HDIM;

  float* ws    = (float*)d_ws;
  float* eacc  = ws;                          // V*H  segment-sum accumulator
  float* left  = ws + (size_t)V * HDIM;       // V*H
  float* right = ws + 2 * (size_t)V * HDIM;   // V*H

  const long zn = (long)V * HDIM;
  zero_kernel<<<(int)((zn + 255) / 256), 256, 0, stream>>>(eacc, zn);

  const int tilesV = (V + 15) / 16;
  node_lr_kernel<<<(tilesV + 7) / 8, 256, 0, stream>>>(nf, w_l, b_l, w_r, b_r, left, right, V);

  const int tilesE = (E + 15) / 16;
  edge_kernel<<<(tilesE + 7) / 8, 256, 0, stream>>>(ef, src, dst, left, right,
                                                    w_e2n, b_e2n, w_e2e, b_e2e,
                                                    w_upd_e, b_upd_e, eacc, new_edge, E);

  node_update_kernel<<<(tilesV + 7) / 8, 256, 0, stream>>>(nf, w_n2n, b_n2n,
                                                           w_upd_n, b_upd_n, eacc, new_node, V);
}